// GCN_56435870269978
// MI455X (gfx1250) — compile-verified
//
#include <hip/hip_runtime.h>

// ---------------------------------------------------------------------------
// GCN forward on gfx1250 (MI455X), wave32.
//   h1 = relu( Agg( x @ W1 ) + b1 )
//   h2 = relu( Agg( h1 @ W2 ) + b2 )
//   out = h2 @ Wout + bout
// Agg(hw)[i] = dinv[i]^2*hw[i] + sum_{e: dst=i} dinv[src]*dinv[i]*hw[src]
//
// Roofline: edge aggregation dominates (~240M f32 atomics on L2-resident
// 40MB buffers); the 5 GFLOP of GEMM work is nearly free, so GEMMs use the
// exact-precision V_WMMA_F32_16X16X4_F32 path with both operands staged in
// LDS (conflict-free stride-210 layout, ds_load_b64 inner loop).
// ---------------------------------------------------------------------------

typedef float v2f __attribute__((ext_vector_type(2)));
typedef float v8f __attribute__((ext_vector_type(8)));

#define N_NODES 50000
#define N_EDGES 800000
#define F_IN    128
#define H1      200
#define H1P     208   // padded to multiple of 16
#define H2      100
#define H2P     112
#define NC      40
#define NCP     48

#define KP      210   // LDS row/col stride in floats: conflict-free, keeps b64 alignment

// ---------------- degree / norm ----------------
__global__ void k_deg_init(float* __restrict__ deg, int n) {
    int i = blockIdx.x * blockDim.x + threadIdx.x;
    if (i < n) deg[i] = 1.0f;             // self-loop contributes 1
}

__global__ void k_deg_accum(const int* __restrict__ dst, float* __restrict__ deg, int e) {
    int i = blockIdx.x * blockDim.x + threadIdx.x;
    if (i < e) atomicAdd(&deg[dst[i]], 1.0f);
}

__global__ void k_rsqrt_inplace(float* __restrict__ v, int n) {
    int i = blockIdx.x * blockDim.x + threadIdx.x;
    if (i < n) v[i] = rsqrtf(v[i]);
}

// ---------------- WMMA f32 GEMM ----------------
// C tile = op(A[M x K, lda]) @ B[K x Nact].  block = 128 threads (4 wave32s);
// grid = (M/16, ceil(Npad/64)). A tile (16 x K, optional fused ReLU) and the
// block's 64-column B slice (transposed to column-major) are staged in LDS;
// the inner loop is pure ds_load_b64 + v_wmma. Columns >= Nact are staged as
// zeros so padded output columns come out exactly 0.
__global__ void __launch_bounds__(128)
k_gemm_wmma_f32(const float* __restrict__ A, int lda,
                const float* __restrict__ B, int Nact,
                float* __restrict__ C, int ldc, int Nstore,
                int K, int reluA, const float* __restrict__ bias)
{
    __shared__ float As[16 * KP];   // [row][k]
    __shared__ float Bs[64 * KP];   // [col][k]  (transposed from global)

    const int m_base = blockIdx.x * 16;
    const int nblk   = blockIdx.y * 64;
    const int tid    = threadIdx.x;

    // Stage A tile (fused ReLU for layers 2/3).
    for (int idx = tid; idx < 16 * K; idx += 128) {
        int r = idx / K;
        int c = idx - r * K;
        float v = A[(size_t)(m_base + r) * lda + c];
        if (reluA) v = fmaxf(v, 0.0f);
        As[r * KP + c] = v;
    }
    // Stage B slice, transposed: Bs[c][k] = B[k][nblk+c] (0 beyond Nact).
    for (int idx = tid; idx < 64 * K; idx += 128) {
        int k = idx >> 6;
        int c = idx & 63;
        int col = nblk + c;
        float v = (col < Nact) ? B[(size_t)k * Nact + col] : 0.0f;
        Bs[c * KP + k] = v;
    }
    __syncthreads();

    const int wave = tid >> 5;
    const int lane = tid & 31;
    const int mloc = lane & 15;            // A row / B-C column within tile
    const int kb   = (lane >> 4) << 1;     // K sub-offset: lanes 0-15 -> 0, 16-31 -> 2
    const int ncol = nblk + wave * 16 + mloc;

    const float* ap = &As[mloc * KP + kb];
    const float* bp = &Bs[(wave * 16 + mloc) * KP + kb];

    v8f acc = {0.f, 0.f, 0.f, 0.f, 0.f, 0.f, 0.f, 0.f};

    for (int k = 0; k < K; k += 4) {
        v2f a, b;
        // A 16x4 f32 layout: lane m, VGPR v holds K = (lane/16)*2 + v
        a.x = ap[k];
        a.y = ap[k + 1];
        // B 4x16 f32 layout: lane n, VGPR v holds K = (lane/16)*2 + v
        b.x = bp[k];
        b.y = bp[k + 1];
        acc = __builtin_amdgcn_wmma_f32_16x16x4_f32(
                  false, a, false, b, (short)0, acc, false, false);
    }

    float bv = (bias != nullptr && ncol < Nact) ? bias[ncol] : 0.0f;
    if (ncol < Nstore) {
        // C/D layout: VGPR g -> row m_base+g (lanes 0-15) / m_base+8+g (16-31)
        const int rbase = m_base + ((lane >> 4) << 3);
        #pragma unroll
        for (int g = 0; g < 8; ++g)
            C[(size_t)(rbase + g) * ldc + ncol] = acc[g] + bv;
    }
}

// ---------------- aggregation ----------------
// agg[i][f] = (f<Fact ? b[f] : 0) + dinv[i]^2 * hw[i][f]   (self-loop + bias)
__global__ void k_agg_init(const float* __restrict__ hw, const float* __restrict__ dinv,
                           const float* __restrict__ bias, float* __restrict__ agg,
                           int Fp, int Fact, int total)
{
    int i = blockIdx.x * blockDim.x + threadIdx.x;
    if (i >= total) return;
    int node = i / Fp;
    int f    = i - node * Fp;
    float d  = dinv[node];
    float bb = (f < Fact) ? bias[f] : 0.0f;
    agg[i] = bb + d * d * hw[i];
}

// One wave per edge (grid-stride); lanes stride the feature dimension.
__global__ void k_agg_edges(const int* __restrict__ src, const int* __restrict__ dst,
                            const float* __restrict__ dinv, const float* __restrict__ hw,
                            float* __restrict__ agg, int Fp, int Fact, int e)
{
    int wid    = (blockIdx.x * blockDim.x + threadIdx.x) >> 5;
    int lane   = threadIdx.x & 31;
    int nwaves = (gridDim.x * blockDim.x) >> 5;
    for (int ed = wid; ed < e; ed += nwaves) {
        int s = src[ed];
        int d = dst[ed];
        float w = dinv[s] * dinv[d];
        const float* hs = hw  + (size_t)s * Fp;
        float*       ad = agg + (size_t)d * Fp;
        for (int f = lane; f < Fact; f += 32)
            atomicAdd(&ad[f], w * hs[f]);
    }
}

// ---------------------------------------------------------------------------
extern "C" void kernel_launch(void* const* d_in, const int* in_sizes, int n_in,
                              void* d_out, int out_size, void* d_ws, size_t ws_size,
                              hipStream_t stream)
{
    const float* x    = (const float*)d_in[0];
    const int*   ei   = (const int*)d_in[1];     // (2, E): row 0 = src, row 1 = dst
    const float* W1   = (const float*)d_in[2];
    const float* b1   = (const float*)d_in[3];
    const float* W2   = (const float*)d_in[4];
    const float* b2   = (const float*)d_in[5];
    const float* Wout = (const float*)d_in[6];
    const float* bout = (const float*)d_in[7];
    float*       out  = (float*)d_out;

    const int* srcI = ei;
    const int* dstI = ei + N_EDGES;

    // Workspace layout (floats)
    float* ws   = (float*)d_ws;
    float* dinv = ws;                                    // 50000
    float* bufA = ws + 50176;                            // 50000*208 (hw)
    float* bufB = bufA + (size_t)N_NODES * H1P;          // 50000*208 (agg/h)

    // ---- symmetric normalization ----
    k_deg_init     <<<(N_NODES + 255) / 256, 256, 0, stream>>>(dinv, N_NODES);
    k_deg_accum    <<<(N_EDGES + 255) / 256, 256, 0, stream>>>(dstI, dinv, N_EDGES);
    k_rsqrt_inplace<<<(N_NODES + 255) / 256, 256, 0, stream>>>(dinv, N_NODES);

    // ---- layer 1: hw1 = x @ W1 ; agg1 = Agg(hw1) + b1 ----
    k_gemm_wmma_f32<<<dim3(N_NODES / 16, (H1P + 63) / 64), 128, 0, stream>>>(
        x, F_IN, W1, H1, bufA, H1P, H1P, F_IN, /*relu=*/0, /*bias=*/nullptr);
    {
        int total = N_NODES * H1P;
        k_agg_init<<<(total + 255) / 256, 256, 0, stream>>>(bufA, dinv, b1, bufB,
                                                            H1P, H1, total);
    }
    k_agg_edges<<<4096, 256, 0, stream>>>(srcI, dstI, dinv, bufA, bufB, H1P, H1, N_EDGES);

    // ---- layer 2: hw2 = relu(agg1) @ W2 ; agg2 = Agg(hw2) + b2 ----
    k_gemm_wmma_f32<<<dim3(N_NODES / 16, (H2P + 63) / 64), 128, 0, stream>>>(
        bufB, H1P, W2, H2, bufA, H2P, H2P, H1, /*relu=*/1, /*bias=*/nullptr);
    {
        int total = N_NODES * H2P;
        k_agg_init<<<(total + 255) / 256, 256, 0, stream>>>(bufA, dinv, b2, bufB,
                                                            H2P, H2, total);
    }
    k_agg_edges<<<4096, 256, 0, stream>>>(srcI, dstI, dinv, bufA, bufB, H2P, H2, N_EDGES);

    // ---- head: out = relu(agg2) @ Wout + bout (compact ldc = 40) ----
    k_gemm_wmma_f32<<<dim3(N_NODES / 16, (NCP + 63) / 64), 128, 0, stream>>>(
        bufB, H2P, Wout, NC, out, NC, NC, H2, /*relu=*/1, /*bias=*/bout);
}